// AttentionLegalbasis_89103391523145
// MI455X (gfx1250) — compile-verified
//
#include <hip/hip_runtime.h>
#include <math.h>

typedef __attribute__((ext_vector_type(2))) float v2f;
typedef __attribute__((ext_vector_type(8))) float v8f;

#define BATCH 32
#define SEQ   4096
#define HID   768
#define WAVES_PER_BATCH 32          // 1024 waves on the device
#define BLOCKS_PER_BATCH 8          // 4 waves (128 threads) per block
#define WAVE_ROWS (SEQ / WAVES_PER_BATCH)   // 128 rows per wave
#define GROUPS (WAVE_ROWS / 16)             // 8 groups of 16 rows
#define REC_STRIDE 772              // floats per partial record (16B aligned)

// ---------------------------------------------------------------------------
// Pass 1: per-wave online-softmax over a 128-row slice of one batch.
// Scores via V_WMMA_F32_16X16X4_F32 (A = 16 rows x 4 K of hidden,
// B = query chunk replicated over all 16 columns -> C columns all equal the
// 16 scores, broadcast to every lane nearly for free).
// ---------------------------------------------------------------------------
__global__ __launch_bounds__(128) void attn_pass1(
    const float* __restrict__ hidden,   // [BATCH, SEQ, HID]
    const float* __restrict__ q,        // [HID]
    float* __restrict__ ws)             // partial records
{
    __shared__ float qs[HID];
    const int tid = threadIdx.x;
    for (int i = tid; i < HID; i += 128) qs[i] = q[i];
    __syncthreads();

    const int b    = blockIdx.y;
    const int lane = tid & 31;
    const int w    = blockIdx.x * 4 + (tid >> 5);   // wave index within batch
    const int row0 = w * WAVE_ROWS;

    const float* hb = hidden + (size_t)b * SEQ * HID;

    const int mrow = lane & 15;          // M row this lane supplies for A
    const int koff = (lane >> 4) * 2;    // lanes 16-31 hold K+2,K+3

    float m = -INFINITY;
    float l = 0.f;
    float acc[24];                       // h = blk*128 + lane*4 + i
#pragma unroll
    for (int t = 0; t < 24; ++t) acc[t] = 0.f;

    for (int g = 0; g < GROUPS; ++g) {
        const int rbase = row0 + g * 16;
        const float* aptr = hb + (size_t)(rbase + mrow) * HID + koff;

        // ---- scores for 16 rows: 192 chained f32 WMMAs (K=768) ----
        v8f c = {};
#pragma unroll 8
        for (int j = 0; j < HID / 4; ++j) {
            v2f a  = *(const v2f*)(aptr + 4 * j);          // global_load_b64
            v2f bq = *(const v2f*)(qs + 4 * j + koff);     // ds_load_b64 (bcast)
            c = __builtin_amdgcn_wmma_f32_16x16x4_f32(
                    false, a, false, bq, (short)0, c, false, false);
        }

        // C columns are replicated: lane holds scores m = r + (lane>=16 ? 8:0).
        // One xor-16 shuffle per register gives every lane all 16 scores.
        float sc[16];
#pragma unroll
        for (int r = 0; r < 8; ++r) {
            float mine  = c[r];
            float other = __shfl_xor(mine, 16, 32);
            sc[r]     = (lane < 16) ? mine : other;
            sc[r + 8] = (lane < 16) ? other : mine;
        }

        // ---- online softmax update ----
        float newm = m;
#pragma unroll
        for (int r = 0; r < 16; ++r) newm = fmaxf(newm, sc[r]);
        const float scale = __expf(m - newm);   // exp(-inf)=0 on first group
        m = newm;
        l *= scale;
#pragma unroll
        for (int t = 0; t < 24; ++t) acc[t] *= scale;

        float wgt[16];
#pragma unroll
        for (int r = 0; r < 16; ++r) {
            wgt[r] = __expf(sc[r] - newm);
            l += wgt[r];
        }

        // ---- weighted accumulation; rows are WGP$/L2 resident ----
#pragma unroll 4
        for (int r = 0; r < 16; ++r) {
            const float* hr = hb + (size_t)(rbase + r) * HID + lane * 4;
#pragma unroll
            for (int blk = 0; blk < 6; ++blk) {
                float4 hv = *(const float4*)(hr + blk * 128); // b128, coalesced
                acc[blk * 4 + 0] = fmaf(wgt[r], hv.x, acc[blk * 4 + 0]);
                acc[blk * 4 + 1] = fmaf(wgt[r], hv.y, acc[blk * 4 + 1]);
                acc[blk * 4 + 2] = fmaf(wgt[r], hv.z, acc[blk * 4 + 2]);
                acc[blk * 4 + 3] = fmaf(wgt[r], hv.w, acc[blk * 4 + 3]);
            }
        }
    }

    // ---- write partial record: [m, l, pad, pad, acc[768]] ----
    float* rec = ws + (size_t)(b * WAVES_PER_BATCH + w) * REC_STRIDE;
    if (lane == 0) { rec[0] = m; rec[1] = l; }
#pragma unroll
    for (int blk = 0; blk < 6; ++blk) {
        *(float4*)(rec + 4 + blk * 128 + lane * 4) =
            make_float4(acc[blk * 4 + 0], acc[blk * 4 + 1],
                        acc[blk * 4 + 2], acc[blk * 4 + 3]);
    }
}

// ---------------------------------------------------------------------------
// Pass 2: merge the 32 partials of each batch (log-sum-exp combine).
// ---------------------------------------------------------------------------
__global__ __launch_bounds__(256) void attn_pass2(
    const float* __restrict__ ws, float* __restrict__ out)
{
    const int b = blockIdx.x;
    const float* base = ws + (size_t)b * WAVES_PER_BATCH * REC_STRIDE;

    float M = -INFINITY;
#pragma unroll
    for (int i = 0; i < WAVES_PER_BATCH; ++i)
        M = fmaxf(M, base[i * REC_STRIDE]);

    float e[WAVES_PER_BATCH];
    float L = 0.f;
#pragma unroll
    for (int i = 0; i < WAVES_PER_BATCH; ++i) {
        e[i] = __expf(base[i * REC_STRIDE] - M);
        L += base[i * REC_STRIDE + 1] * e[i];
    }
    const float inv = 1.f / L;

    for (int h = threadIdx.x; h < HID; h += 256) {
        float s = 0.f;
#pragma unroll
        for (int i = 0; i < WAVES_PER_BATCH; ++i)
            s += base[i * REC_STRIDE + 4 + h] * e[i];
        out[(size_t)b * HID + h] = s * inv;
    }
}

extern "C" void kernel_launch(void* const* d_in, const int* in_sizes, int n_in,
                              void* d_out, int out_size, void* d_ws, size_t ws_size,
                              hipStream_t stream)
{
    const float* hidden = (const float*)d_in[0];
    const float* q      = (const float*)d_in[1];
    float* out          = (float*)d_out;
    float* ws           = (float*)d_ws;

    dim3 grid1(BLOCKS_PER_BATCH, BATCH);
    attn_pass1<<<grid1, 128, 0, stream>>>(hidden, q, ws);
    attn_pass2<<<BATCH, 256, 0, stream>>>(ws, out);
}